// SubArrayHybridBeamformingNet_9448928051418
// MI455X (gfx1250) — compile-verified
//
#include <hip/hip_runtime.h>
#include <math.h>

typedef __attribute__((ext_vector_type(2))) float v2f;
typedef __attribute__((ext_vector_type(4))) float v4f;
typedef __attribute__((ext_vector_type(8))) float v8f;

static constexpr int  B_SZ  = 1024;
static constexpr int  FEATS = 512;
static constexpr int  M_SZ  = 256;
static constexpr int  KK    = 256;     // K*K = 16*16
static constexpr int  NCOL  = 768;     // theta(256) | Dr(256) | Di(256)

static constexpr long A_ELEMS = (long)B_SZ * M_SZ * M_SZ;   // 67108864 per matrix
static constexpr long DR_OFF  = 2 * A_ELEMS;
static constexpr long DI_OFF  = DR_OFF + (long)B_SZ * KK;
static constexpr long C_OFF   = DI_OFF + (long)B_SZ * KK;

// --------------------------------------------------------------------------
// Kernel 1: fused triple GEMM  out[1024 x 768] = x[1024x512] @ [Wθ|Wr|Wi]^T
// One 16x16 output tile per wave32, K-loop of V_WMMA_F32_16X16X4_F32.
//   A fragment (16x4 f32): lane = (k/2)*16 + m, vgpr = k%2  -> float2 load
//   B fragment (4x16 f32): lane = (k/2)*16 + n, vgpr = k%2  -> float2 load of W[n][k..k+1]
//   C/D: vgpr r -> row (r + 8*(lane/16)), col = lane%16
// Epilogue: +bias; group 0 -> sincos into ws; groups 1/2 -> raw D into d_out.
// --------------------------------------------------------------------------
__global__ __launch_bounds__(256)
void gemm_wmma_kernel(const float* __restrict__ x,
                      const float* __restrict__ Wt, const float* __restrict__ bt,
                      const float* __restrict__ Wr, const float* __restrict__ br,
                      const float* __restrict__ Wi, const float* __restrict__ bi,
                      float* __restrict__ out,
                      float* __restrict__ ar, float* __restrict__ ai)
{
    const int lane = threadIdx.x & 31;
    const int wid  = blockIdx.x * 8 + (threadIdx.x >> 5);   // 3072 wave-tiles total
    const int tile_row = wid / (NCOL / 16);                 // 0..63
    const int tile_col = wid % (NCOL / 16);                 // 0..47
    const int hi = lane >> 4;                               // K-pair select
    const int lm = lane & 15;                               // m (for A) / n (for B)

    const int col_base  = tile_col * 16;
    const int group     = col_base >> 8;                    // 0=theta 1=Dr 2=Di
    const int lcol_base = col_base & 255;
    const float* W    = (group == 0) ? Wt : (group == 1) ? Wr : Wi;
    const float* bias = (group == 0) ? bt : (group == 1) ? br : bi;

    const float* ap = x + (size_t)(tile_row * 16 + lm) * FEATS + 2 * hi;
    const float* bp = W + (size_t)(lcol_base + lm) * FEATS + 2 * hi;

    v8f acc = {};
#pragma unroll 8
    for (int k0 = 0; k0 < FEATS; k0 += 4) {
        v2f a = *(const v2f*)(ap + k0);
        v2f b = *(const v2f*)(bp + k0);
        acc = __builtin_amdgcn_wmma_f32_16x16x4_f32(
            /*neg_a=*/false, a, /*neg_b=*/false, b,
            /*c_mod=*/(short)0, acc, /*reuse_a=*/false, /*reuse_b=*/false);
    }

    const int   col = lcol_base + lm;
    const float bv  = bias[col];

#pragma unroll
    for (int r = 0; r < 8; ++r) {
        const int   row = tile_row * 16 + r + 8 * hi;       // batch index
        const float v   = acc[r] + bv;
        if (group == 0) {
            float s, c;
            __sincosf(v, &s, &c);
            ar[(size_t)row * M_SZ + col] = c;
            ai[(size_t)row * M_SZ + col] = s;
        } else if (group == 1) {
            out[DR_OFF + (size_t)row * KK + col] = v;
        } else {
            out[DI_OFF + (size_t)row * KK + col] = v;
        }
    }
}

// --------------------------------------------------------------------------
// Kernel 2: per-batch Frobenius norm + in-place scale of D_real / D_imag.
// ||V||_F^2 = 16 * sum(Dr^2 + Di^2)  (since cos^2+sin^2 = 1 and C selects
// each block row 16 times).  scale = sqrt(10) / sqrt(16 * sum).
// --------------------------------------------------------------------------
__global__ __launch_bounds__(256)
void norm_kernel(float* __restrict__ out)
{
    __shared__ float red[256];
    const int b = blockIdx.x;
    const int t = threadIdx.x;
    const float dr = out[DR_OFF + (size_t)b * KK + t];
    const float di = out[DI_OFF + (size_t)b * KK + t];
    red[t] = dr * dr + di * di;
    __syncthreads();
#pragma unroll
    for (int s = 128; s > 0; s >>= 1) {
        if (t < s) red[t] += red[t + s];
        __syncthreads();
    }
    const float scale = sqrtf(10.0f) * __frsqrt_rn(16.0f * red[0]);
    out[DR_OFF + (size_t)b * KK + t] = dr * scale;
    out[DI_OFF + (size_t)b * KK + t] = di * scale;
}

// --------------------------------------------------------------------------
// Kernel 3: stream-fill A_real / A_imag (537 MB, bandwidth floor of the
// problem) with non-temporal float4 stores; insert diagonal cos/sin.
// One thread per 4-element chunk: 2 * 1024 * 256 * 64 = 33,554,432 threads.
// --------------------------------------------------------------------------
__global__ __launch_bounds__(256)
void afill_kernel(const float* __restrict__ ar, const float* __restrict__ ai,
                  float* __restrict__ out)
{
    const unsigned gid = blockIdx.x * 256u + threadIdx.x;
    const unsigned mat = gid >> 24;            // 0 = A_real, 1 = A_imag
    const unsigned rem = gid & 0xFFFFFFu;
    const unsigned bm  = rem >> 6;             // b*256 + m
    const unsigned m   = bm & 255u;
    const unsigned c   = rem & 63u;            // chunk of 4 columns
    v4f val = {0.f, 0.f, 0.f, 0.f};
    if ((m >> 2) == c) {                       // this chunk holds the diagonal
        val[m & 3u] = mat ? ai[bm] : ar[bm];
    }
    float* dst = out + (mat ? A_ELEMS : 0) + (size_t)bm * 256 + (size_t)c * 4;
    __builtin_nontemporal_store(val, (v4f*)dst);
}

// --------------------------------------------------------------------------
// Kernel 4: pass-through copy of C [256 x 16].
// --------------------------------------------------------------------------
__global__ __launch_bounds__(256)
void ccopy_kernel(const float* __restrict__ Cin, float* __restrict__ out)
{
    const int i = blockIdx.x * 256 + threadIdx.x;
    out[C_OFF + i] = Cin[i];
}

extern "C" void kernel_launch(void* const* d_in, const int* in_sizes, int n_in,
                              void* d_out, int out_size, void* d_ws, size_t ws_size,
                              hipStream_t stream)
{
    const float* x  = (const float*)d_in[0];
    const float* Wt = (const float*)d_in[1];
    const float* bt = (const float*)d_in[2];
    const float* Wr = (const float*)d_in[3];
    const float* br = (const float*)d_in[4];
    const float* Wi = (const float*)d_in[5];
    const float* bi = (const float*)d_in[6];
    const float* C  = (const float*)d_in[7];
    float* out = (float*)d_out;

    float* ar = (float*)d_ws;                        // [1024*256] cos(theta)
    float* ai = ar + (size_t)B_SZ * M_SZ;            // [1024*256] sin(theta)

    // 3072 wave-tiles / 8 waves per 256-thread block = 384 blocks
    gemm_wmma_kernel<<<384, 256, 0, stream>>>(x, Wt, bt, Wr, br, Wi, bi, out, ar, ai);
    norm_kernel<<<B_SZ, 256, 0, stream>>>(out);
    afill_kernel<<<131072, 256, 0, stream>>>(ar, ai, out);   // 33.5M threads
    ccopy_kernel<<<16, 256, 0, stream>>>(C, out);
}